// REN_11931419148886
// MI455X (gfx1250) — compile-verified
//
#include <hip/hip_runtime.h>
#include <math.h>

// ---------------- problem constants ----------------
#define DXC 512
#define LC  512
#define DINC 256
#define DOUTC 256
#define BC 8192
#define HN 1536              // 2*DX + L
#define EPSF 0.001f

typedef float v2f __attribute__((ext_vector_type(2)));
typedef float v8f __attribute__((ext_vector_type(8)));
typedef unsigned int u32x4 __attribute__((ext_vector_type(4)));
typedef int i32x4v __attribute__((ext_vector_type(4)));
typedef int i32x8v __attribute__((ext_vector_type(8)));

// ---------------- WMMA f32 16x16x4 helpers ----------------
// A fragment (16x4 f32): VGPR0 = K+0 (lanes 0-15) / K+2 (lanes 16-31), VGPR1 = K+1 / K+3.
__device__ __forceinline__ v2f frag_rm(const float* __restrict__ p, int ld, int row0, int k, int lane) {
    int r  = lane & 15;
    int kk = k + ((lane >> 4) << 1);
    const float* q = p + (size_t)(row0 + r) * ld + kk;
    v2f f; f.x = q[0]; f.y = q[1];
    return f;
}
// element A[row,k] = p[k*ld + row]; also the B fragment for row-major B in C = A@B.
__device__ __forceinline__ v2f frag_cm(const float* __restrict__ p, int ld, int row0, int k, int lane) {
    int r  = lane & 15;
    int kk = k + ((lane >> 4) << 1);
    v2f f;
    f.x = p[(size_t)kk * ld + row0 + r];
    f.y = p[(size_t)(kk + 1) * ld + row0 + r];
    return f;
}
__device__ __forceinline__ v8f wmma4(v2f a, v2f b, v8f c) {
    return __builtin_amdgcn_wmma_f32_16x16x4_f32(false, a, false, b, (short)0, c, false, false);
}
__device__ __forceinline__ void store_tile(float* __restrict__ C, int ldc, int m0, int n0, int lane, v8f acc) {
    int r = lane & 15, h = lane >> 4;
#pragma unroll
    for (int i = 0; i < 8; ++i)
        C[(size_t)(m0 + h * 8 + i) * ldc + n0 + r] = acc[i];
}
__device__ __forceinline__ void store4(float* __restrict__ C, int ldc, int m0, int n0, int lane, const v8f* acc) {
#pragma unroll
    for (int t = 0; t < 4; ++t) store_tile(C, ldc, m0, n0 + 16 * t, lane, acc[t]);
}

// acc[0..3] += A[m0:+16, :K] @ W[n0+16t:+16, :K]^T  (both row-major; A-frag reused 4x)
__device__ __forceinline__ void mm_rm(v8f* acc, const float* __restrict__ A, int lda,
                                      const float* __restrict__ W, int ldw,
                                      int m0, int n0, int K, int lane, float wsign) {
    for (int k = 0; k < K; k += 4) {
        v2f a = frag_rm(A, lda, m0, k, lane);
#pragma unroll
        for (int t = 0; t < 4; ++t) {
            v2f b = frag_rm(W, ldw, n0 + 16 * t, k, lane);
            b.x *= wsign; b.y *= wsign;
            acc[t] = wmma4(a, b, acc[t]);
        }
    }
}
// acc[0..3] += A[m0:+16, :K] @ B[:K, n0+16t:+16]   (row-major A and B)
__device__ __forceinline__ void mm_nn(v8f* acc, const float* __restrict__ A, int lda,
                                      const float* __restrict__ B, int ldb,
                                      int m0, int n0, int K, int lane) {
    for (int k = 0; k < K; k += 4) {
        v2f a = frag_rm(A, lda, m0, k, lane);
#pragma unroll
        for (int t = 0; t < 4; ++t)
            acc[t] = wmma4(a, frag_cm(B, ldb, n0 + 16 * t, k, lane), acc[t]);
    }
}

// ---------------- TDM helper: stage one 512-float row of H into LDS ----------------
// D# per CDNA5 ISA: group0 = {count=1 | lds_addr | global_addr | type=2},
// group1 = {data_size=4B, tensor_dim0=512, tensor_dim1=1, tile 512x1, dim0_stride=1536}.
__device__ __forceinline__ void tdm_load_row(const float* __restrict__ gsrc, unsigned lds_byte_addr) {
    unsigned long long ga = (unsigned long long)(size_t)gsrc;
    u32x4 g0;
    g0[0] = 1u;                                            // count = 1 valid descriptor
    g0[1] = lds_byte_addr;                                 // lds_addr
    g0[2] = (unsigned)(ga & 0xffffffffu);                  // global_addr[31:0]
    g0[3] = (unsigned)((ga >> 32) & 0x01ffffffu) | (2u << 30); // global_addr[56:32] | type=2
    i32x8v g1;
    g1[0] = 0x00020000;                                    // data_size = 2 (4 bytes)
    g1[1] = (int)(512u << 16);                             // tensor_dim0 = 512
    g1[2] = (int)(1u << 16);                               // tensor_dim1 = 1
    g1[3] = (int)(512u << 16);                             // tile_dim0 = 512
    g1[4] = 1;                                             // tile_dim1 = 1, tile_dim2 = 0
    g1[5] = 1536;                                          // tensor_dim0_stride = HN
    g1[6] = 0;
    g1[7] = 0;
    i32x4v z4 = {0, 0, 0, 0};
    i32x8v z8 = {0, 0, 0, 0, 0, 0, 0, 0};
    __builtin_amdgcn_tensor_load_to_lds(g0, g1, z4, z4, z8, 0);
}

// ---------------- kernels ----------------

// H = X^T X + eps*I (1536x1536).  grid = (HN/64, HN/64), block = 128 (4 waves, 16x64 strip/wave).
__global__ __launch_bounds__(128) void k_syrk(const float* __restrict__ X, float* __restrict__ H) {
    const int lane = threadIdx.x & 31, wave = threadIdx.x >> 5;
    const int n0 = blockIdx.x * 64;
    const int m0 = blockIdx.y * 64 + wave * 16;
    v8f acc[4] = {};
    for (int k = 0; k < HN; k += 4) {
        v2f a = frag_cm(X, HN, m0, k, lane);         // A[m,k] = X[k,m]
#pragma unroll
        for (int t = 0; t < 4; ++t)
            acc[t] = wmma4(a, frag_cm(X, HN, n0 + 16 * t, k, lane), acc[t]);
    }
    int r = lane & 15, h = lane >> 4;
#pragma unroll
    for (int t = 0; t < 4; ++t)
#pragma unroll
        for (int i = 0; i < 8; ++i) {
            int gm = m0 + h * 8 + i, gn = n0 + 16 * t + r;
            H[(size_t)gm * HN + gn] = acc[t][i] + (gm == gn ? EPSF : 0.0f);
        }
}

__global__ void k_lam(const float* __restrict__ H, float* __restrict__ laminv) {
    int i = blockIdx.x * blockDim.x + threadIdx.x;
    if (i < LC) laminv[i] = 2.0f / H[(size_t)(DXC + i) * HN + DXC + i];
}

__global__ void k_e(const float* __restrict__ H, const float* __restrict__ Y, float* __restrict__ E) {
    int idx = blockIdx.x * blockDim.x + threadIdx.x;
    if (idx >= DXC * DXC) return;
    int i = idx >> 9, j = idx & 511;
    float h11 = H[(size_t)i * HN + j];
    float h33 = H[(size_t)(DXC + LC + i) * HN + DXC + LC + j];
    E[idx] = 0.5f * (h11 + h33 + Y[(size_t)i * DXC + j] - Y[(size_t)j * DXC + i]);
}

__global__ __launch_bounds__(512) void k_norms(const float* __restrict__ E, float* __restrict__ norms) {
    __shared__ float red[512];
    int t = threadIdx.x;
    float rs = 0.f, cs = 0.f;
    for (int j = 0; j < DXC; ++j) {
        rs += fabsf(E[(size_t)t * DXC + j]);
        cs += fabsf(E[(size_t)j * DXC + t]);
    }
    red[t] = rs; __syncthreads();
    for (int s = 256; s > 0; s >>= 1) { if (t < s) red[t] = fmaxf(red[t], red[t + s]); __syncthreads(); }
    float rinf = red[0]; __syncthreads();
    red[t] = cs; __syncthreads();
    for (int s = 256; s > 0; s >>= 1) { if (t < s) red[t] = fmaxf(red[t], red[t + s]); __syncthreads(); }
    if (t == 0) norms[0] = 1.0f / (rinf * red[0]);
}

__global__ void k_xinit(const float* __restrict__ E, const float* __restrict__ norms, float* __restrict__ X0) {
    int idx = blockIdx.x * blockDim.x + threadIdx.x;
    if (idx >= DXC * DXC) return;
    int i = idx >> 9, j = idx & 511;
    X0[idx] = norms[0] * E[(size_t)j * DXC + i];
}

// T = E @ Xk  (512^3).  grid = (8, 8).
__global__ __launch_bounds__(128) void k_gemm_nn512(const float* __restrict__ A, const float* __restrict__ B,
                                                    float* __restrict__ C) {
    const int lane = threadIdx.x & 31, wave = threadIdx.x >> 5;
    const int n0 = blockIdx.x * 64;
    const int m0 = blockIdx.y * 64 + wave * 16;
    v8f acc[4] = {};
    mm_nn(acc, A, DXC, B, DXC, m0, n0, DXC, lane);
    store4(C, DXC, m0, n0, lane, acc);
}

// Xn = 2*Xk - Xk @ T   (Newton-Schulz update)
__global__ __launch_bounds__(128) void k_ns(const float* __restrict__ Xk, const float* __restrict__ T,
                                            float* __restrict__ Xn) {
    const int lane = threadIdx.x & 31, wave = threadIdx.x >> 5;
    const int n0 = blockIdx.x * 64;
    const int m0 = blockIdx.y * 64 + wave * 16;
    v8f acc[4] = {};
    mm_nn(acc, Xk, DXC, T, DXC, m0, n0, DXC, lane);
    int r = lane & 15, h = lane >> 4;
#pragma unroll
    for (int t = 0; t < 4; ++t)
#pragma unroll
        for (int i = 0; i < 8; ++i) {
            size_t off = (size_t)(m0 + h * 8 + i) * DXC + n0 + 16 * t + r;
            Xn[off] = 2.0f * Xk[off] - acc[t][i];
        }
}

// a = u @ D12^T - x0 @ H21^T   (B x 512).  grid = (8, B/64).
__global__ __launch_bounds__(128) void k_a(const float* __restrict__ u, const float* __restrict__ x0,
                                           const float* __restrict__ D12, const float* __restrict__ H,
                                           float* __restrict__ aout) {
    const int lane = threadIdx.x & 31, wave = threadIdx.x >> 5;
    const int n0 = blockIdx.x * 64;
    const int m0 = blockIdx.y * 64 + wave * 16;
    v8f acc[4] = {};
    mm_rm(acc, u, DINC, D12, DINC, m0, n0, DINC, lane, 1.0f);
    mm_rm(acc, x0, DXC, H + (size_t)DXC * HN, HN, m0, n0, DXC, lane, -1.0f);  // C1 = -H21
    store4(aout, DXC, m0, n0, lane, acc);
}

// Sequential triangular scan: w[:,i] = tanh((a[:,i] - sum_{j<i} H22[i,j] w[:,j]) / Lam[i]).
// 64 batch rows/workgroup, w tile in LDS; the H22 row is double-buffered into LDS by the
// Tensor Data Mover (one issue per workgroup), synchronized with s_wait_tensorcnt + barrier.
#define WST 513
__global__ __launch_bounds__(512) void k_scan(const float* __restrict__ a, const float* __restrict__ H,
                                              const float* __restrict__ laminv, float* __restrict__ w) {
    extern __shared__ float smem[];
    float* wt   = smem;                       // [64][WST]
    float* part = smem + 64 * WST;            // [64][8]
    float* drow = part + 64 * 8;              // [2][512] H22-row staging buffers
    const int tid = threadIdx.x;
    const int r = tid >> 3;
    const int s = tid & 7;
    const size_t b0 = (size_t)blockIdx.x * 64;
    const float* H22 = H + (size_t)DXC * HN + DXC;
    const unsigned drow_lds = (unsigned)(size_t)(void*)drow;   // byte addr within LDS space

    if (tid == 0) {
        tdm_load_row(H22, drow_lds);                           // row 0 -> buf 0
        __builtin_amdgcn_s_wait_tensorcnt(0);
    }
    __syncthreads();

    for (int i = 0; i < LC; ++i) {
        if (tid == 0 && i + 1 < LC)                            // prefetch row i+1 -> other buffer
            tdm_load_row(H22 + (size_t)(i + 1) * HN, drow_lds + ((unsigned)(i + 1) & 1u) * 2048u);
        const float* d = drow + (i & 1) * 512;
        float p = 0.f;
        for (int j = s; j < i; j += 8) p -= wt[r * WST + j] * d[j];
        part[r * 8 + s] = p;
        __syncthreads();
        if (s == 0) {
            float tot = part[r*8+0] + part[r*8+1] + part[r*8+2] + part[r*8+3]
                      + part[r*8+4] + part[r*8+5] + part[r*8+6] + part[r*8+7];
            float v = a[(b0 + r) * DXC + i] + tot;
            wt[r * WST + i] = tanhf(v * laminv[i]);
        }
        if (tid == 0 && i + 1 < LC) __builtin_amdgcn_s_wait_tensorcnt(0);
        __syncthreads();                                       // releases staged row i+1 to all waves
    }
    for (int idx = tid; idx < 64 * LC; idx += 512) {
        int rr = idx >> 9, jj = idx & 511;
        w[(b0 + rr) * LC + jj] = wt[rr * WST + jj];
    }
}

// pre = w @ B1^T + u @ B2^T + x0 @ Fm^T   (B x 512)
__global__ __launch_bounds__(128) void k_pre(const float* __restrict__ w, const float* __restrict__ u,
                                             const float* __restrict__ x0, const float* __restrict__ B2,
                                             const float* __restrict__ H, float* __restrict__ pre) {
    const int lane = threadIdx.x & 31, wave = threadIdx.x >> 5;
    const int n0 = blockIdx.x * 64;
    const int m0 = blockIdx.y * 64 + wave * 16;
    v8f acc[4] = {};
    mm_rm(acc, w, LC, H + (size_t)(DXC + LC) * HN + DXC, HN, m0, n0, LC, lane, 1.0f);  // B1 = H32
    mm_rm(acc, u, DINC, B2, DINC, m0, n0, DINC, lane, 1.0f);
    mm_rm(acc, x0, DXC, H + (size_t)(DXC + LC) * HN, HN, m0, n0, DXC, lane, 1.0f);     // Fm = H31
    store4(pre, DXC, m0, n0, lane, acc);
}

// xnew = pre @ Einv   (x_new = pre @ E^{-1} == nn GEMM against row-major Einv)
__global__ __launch_bounds__(128) void k_xnew(const float* __restrict__ pre, const float* __restrict__ Einv,
                                              float* __restrict__ xnew) {
    const int lane = threadIdx.x & 31, wave = threadIdx.x >> 5;
    const int n0 = blockIdx.x * 64;
    const int m0 = blockIdx.y * 64 + wave * 16;
    v8f acc[4] = {};
    mm_nn(acc, pre, DXC, Einv, DXC, m0, n0, DXC, lane);
    store4(xnew, DXC, m0, n0, lane, acc);
}

// y = xnew @ C2^T + w @ D21^T + u @ D22^T   (B x 256) -> d_out
__global__ __launch_bounds__(128) void k_y(const float* __restrict__ xnew, const float* __restrict__ w,
                                           const float* __restrict__ u, const float* __restrict__ C2,
                                           const float* __restrict__ D21, const float* __restrict__ D22,
                                           float* __restrict__ y) {
    const int lane = threadIdx.x & 31, wave = threadIdx.x >> 5;
    const int n0 = blockIdx.x * 64;
    const int m0 = blockIdx.y * 64 + wave * 16;
    v8f acc[4] = {};
    mm_rm(acc, xnew, DXC, C2, DXC, m0, n0, DXC, lane, 1.0f);
    mm_rm(acc, w, LC, D21, LC, m0, n0, LC, lane, 1.0f);
    mm_rm(acc, u, DINC, D22, DINC, m0, n0, DINC, lane, 1.0f);
    store4(y, DOUTC, m0, n0, lane, acc);
}

// ---------------- launcher ----------------
extern "C" void kernel_launch(void* const* d_in, const int* in_sizes, int n_in,
                              void* d_out, int out_size, void* d_ws, size_t ws_size,
                              hipStream_t stream) {
    const float* u   = (const float*)d_in[0];
    const float* x0  = (const float*)d_in[1];
    const float* X   = (const float*)d_in[2];
    const float* Y   = (const float*)d_in[3];
    const float* B2  = (const float*)d_in[4];
    const float* C2  = (const float*)d_in[5];
    const float* D21 = (const float*)d_in[6];
    const float* D22 = (const float*)d_in[7];
    const float* D12 = (const float*)d_in[8];
    float* out = (float*)d_out;

    float* ws = (float*)d_ws;
    float* H      = ws;
    float* E      = H + (size_t)HN * HN;
    float* Xa     = E + (size_t)DXC * DXC;
    float* Xb     = Xa + (size_t)DXC * DXC;
    float* T      = Xb + (size_t)DXC * DXC;
    float* laminv = T + (size_t)DXC * DXC;
    float* norms  = laminv + DXC;
    float* abuf   = norms + 16;
    float* wbuf   = abuf + (size_t)BC * DXC;
    float* pre    = wbuf + (size_t)BC * DXC;
    float* xnew   = pre + (size_t)BC * DXC;

    dim3 blk(128);

    k_syrk<<<dim3(HN / 64, HN / 64), blk, 0, stream>>>(X, H);
    k_lam<<<2, 256, 0, stream>>>(H, laminv);
    k_e<<<(DXC * DXC) / 256, 256, 0, stream>>>(H, Y, E);
    k_norms<<<1, 512, 0, stream>>>(E, norms);
    k_xinit<<<(DXC * DXC) / 256, 256, 0, stream>>>(E, norms, Xa);
    float* Xc = Xa; float* Xn = Xb;
    for (int it = 0; it < 30; ++it) {
        k_gemm_nn512<<<dim3(DXC / 64, DXC / 64), blk, 0, stream>>>(E, Xc, T);
        k_ns<<<dim3(DXC / 64, DXC / 64), blk, 0, stream>>>(Xc, T, Xn);
        float* tmp = Xc; Xc = Xn; Xn = tmp;
    }
    float* Einv = Xc;

    k_a<<<dim3(DXC / 64, BC / 64), blk, 0, stream>>>(u, x0, D12, H, abuf);

    size_t smem = (size_t)(64 * WST + 64 * 8 + 2 * 512) * sizeof(float);
    k_scan<<<BC / 64, 512, smem, stream>>>(abuf, H, laminv, wbuf);

    k_pre<<<dim3(DXC / 64, BC / 64), blk, 0, stream>>>(wbuf, u, x0, B2, H, pre);
    k_xnew<<<dim3(DXC / 64, BC / 64), blk, 0, stream>>>(pre, Einv, xnew);
    k_y<<<dim3(DOUTC / 64, BC / 64), blk, 0, stream>>>(xnew, wbuf, u, C2, D21, D22, out);
}